// PartAdacontLoss_57810259804171
// MI455X (gfx1250) — compile-verified
//
#include <hip/hip_runtime.h>
#include <math.h>

typedef __attribute__((ext_vector_type(2))) float v2f;
typedef __attribute__((ext_vector_type(8))) float v8f;

#define NB 64
#define M  512
#define D  256
#define POS_T 0.7f
#define NEG_T 0.3f
#define BIGF 3.402823466e38f

// workspace layout (bytes)
#define OFF_MASK 0            // uchar[M*M]           = 262144
#define OFF_CNT  262144       // int[M]               = 2048
#define OFF_IHP  264192       // uchar[M]             = 512
#define OFF_IHN  264704       // uchar[M]             = 512
#define OFF_X2   265216       // float[NB*M]          = 131072
#define OFF_PART 396288       // float[4*NB*32]       = 32768
// total ~429 KB

// ---------------------------------------------------------------------------
// Kernel A: per-(i,j) masks + per-row gates (independent of n)
// byte bits: 1=hp, 2=true_hn, 4=pred_hp, 8=pred_hn
// ---------------------------------------------------------------------------
__global__ __launch_bounds__(256)
void mask_kernel(const int* __restrict__ label, const int* __restrict__ ltype,
                 const float* __restrict__ sim,
                 unsigned char* __restrict__ mask, int* __restrict__ cnt_thn,
                 unsigned char* __restrict__ inc_hp, unsigned char* __restrict__ inc_hn)
{
    int i = blockIdx.x;
    int tid = threadIdx.x;
    __shared__ int s_cnt[256];
    __shared__ int s_hp[256];
    __shared__ int s_hn[256];

    int li = label[i];
    int ti = ltype[i];
    int c = 0, ahp = 0, ahn = 0;
    for (int j = tid; j < M; j += 256) {
        int lj = label[j];
        int tj = ltype[j];
        bool hp  = (li == lj);
        bool thn = (!hp) && ((ti > 0) || (ti < tj));
        bool noise = !(hp || thn);
        float s = sim[i * M + j];
        bool php = noise && (s > POS_T);
        bool phn = noise && (s < NEG_T);
        mask[(size_t)i * M + j] = (unsigned char)((hp ? 1 : 0) | (thn ? 2 : 0) |
                                                  (php ? 4 : 0) | (phn ? 8 : 0));
        c += thn ? 1 : 0;
        ahp |= php ? 1 : 0;
        ahn |= phn ? 1 : 0;
    }
    s_cnt[tid] = c; s_hp[tid] = ahp; s_hn[tid] = ahn;
    __syncthreads();
    for (int s2 = 128; s2 > 0; s2 >>= 1) {
        if (tid < s2) {
            s_cnt[tid] += s_cnt[tid + s2];
            s_hp[tid]  |= s_hp[tid + s2];
            s_hn[tid]  |= s_hn[tid + s2];
        }
        __syncthreads();
    }
    if (tid == 0) {
        cnt_thn[i] = s_cnt[0];
        inc_hp[i]  = (unsigned char)s_hp[0];
        inc_hn[i]  = (unsigned char)s_hn[0];
    }
}

// ---------------------------------------------------------------------------
// Kernel B: squared row norms x2[n*M + i] = sum_d F[n,i,d]^2
// ---------------------------------------------------------------------------
__global__ __launch_bounds__(256)
void x2_kernel(const float* __restrict__ feat, float* __restrict__ x2)
{
    int row = blockIdx.x;          // n*M + i
    int tid = threadIdx.x;         // 256 == D
    float v = feat[(size_t)row * D + tid];
    __shared__ float red[256];
    red[tid] = v * v;
    __syncthreads();
    for (int s = 128; s > 0; s >>= 1) {
        if (tid < s) red[tid] += red[tid + s];
        __syncthreads();
    }
    if (tid == 0) x2[row] = red[0];
}

// ---------------------------------------------------------------------------
// Kernel C: WMMA Gram strip -> dist strip in LDS -> row reductions -> losses
// grid = (32 i-tiles, 64 batches), 256 threads = 8 waves
// ---------------------------------------------------------------------------
__global__ __launch_bounds__(256)
void dist_loss_kernel(const float* __restrict__ feat, const float* __restrict__ sim,
                      const unsigned char* __restrict__ mask,
                      const int* __restrict__ cnt_thn,
                      const unsigned char* __restrict__ inc_hp,
                      const unsigned char* __restrict__ inc_hn,
                      const float* __restrict__ x2,
                      float* __restrict__ part)
{
    int it  = blockIdx.x;          // i-tile 0..31
    int n   = blockIdx.y;          // batch 0..63
    int i0  = it * 16;
    int tid = threadIdx.x;

    __shared__ float As[16 * D];   // 16 KB: A strip (rows i0..i0+15)
    __shared__ float Ds[16 * M];   // 32 KB: dist strip
    __shared__ float s_max[16];
    __shared__ float s_min[16];
    __shared__ float sred[256];

    const float* Fn = feat + (size_t)n * M * D;
    const float* x2n = x2 + (size_t)n * M;

    // stage A strip into LDS (coalesced: 256 threads per 256-float row)
    for (int r = 0; r < 16; ++r)
        As[r * D + tid] = Fn[(size_t)(i0 + r) * D + tid];
    __syncthreads();

    int wave  = tid >> 5;          // 0..7
    int lane  = tid & 31;
    int ml    = lane & 15;         // row/col within tile
    int khalf = (lane < 16) ? 0 : 2;
    int mbase = (lane < 16) ? 0 : 8;

    // each wave computes 4 of the 32 j-tiles
    for (int jt = wave; jt < 32; jt += 8) {
        int j0 = jt * 16;
        v8f c = {0.f, 0.f, 0.f, 0.f, 0.f, 0.f, 0.f, 0.f};
        const float* Brow = Fn + (size_t)(j0 + ml) * D;
        for (int k0 = 0; k0 < D; k0 += 4) {
            v2f a, b;
            a.x = As[ml * D + k0 + khalf];
            a.y = As[ml * D + k0 + khalf + 1];
            b.x = Brow[k0 + khalf];
            b.y = Brow[k0 + khalf + 1];
            // D = A(16x4,f32) * B(4x16,f32) + C  -> v_wmma_f32_16x16x4_f32
            c = __builtin_amdgcn_wmma_f32_16x16x4_f32(
                    false, a, false, b, (short)0, c, false, false);
        }
        // Gram tile -> distances, into LDS strip
        float x2j = x2n[j0 + ml];
        for (int r = 0; r < 8; ++r) {
            int mrow = r + mbase;
            float d2 = x2n[i0 + mrow] + x2j - 2.0f * c[r];
            Ds[mrow * M + j0 + ml] = (d2 > 0.f) ? sqrtf(d2) : 0.f;
        }
    }
    __syncthreads();

    // per-row max over hp / min over true_hn (16 threads per row)
    int row = tid >> 4;            // 0..15
    int sub = tid & 15;
    {
        float mx = -BIGF, mn = BIGF;
        const unsigned char* mrow = mask + (size_t)(i0 + row) * M;
        for (int j = sub; j < M; j += 16) {
            float dv = Ds[row * M + j];
            unsigned char mb = mrow[j];
            if (mb & 1) mx = fmaxf(mx, dv);
            if (mb & 2) mn = fminf(mn, dv);
        }
        sred[tid] = mx;
        __syncthreads();
        for (int s = 8; s > 0; s >>= 1) {
            if (sub < s) sred[tid] = fmaxf(sred[tid], sred[tid + s]);
            __syncthreads();
        }
        if (sub == 0) s_max[row] = sred[tid];
        __syncthreads();
        sred[tid] = mn;
        __syncthreads();
        for (int s = 8; s > 0; s >>= 1) {
            if (sub < s) sred[tid] = fminf(sred[tid], sred[tid + s]);
            __syncthreads();
        }
        if (sub == 0) s_min[row] = (cnt_thn[i0 + row] > 0) ? sred[tid] : 0.f;
        __syncthreads();
    }

    // gated losses over the 16x512 strip
    float hp_s = 0.f, hp_c = 0.f, hn_s = 0.f, hn_c = 0.f;
    for (int idx = tid; idx < 16 * M; idx += 256) {
        int r = idx >> 9;
        int j = idx & (M - 1);
        unsigned char mb = mask[(size_t)(i0 + r) * M + j];
        if (mb & 12) {
            float dv = Ds[r * M + j];
            float sv = sim[(size_t)(i0 + r) * M + j];
            if ((mb & 4) && inc_hp[i0 + r]) {
                float l = dv - s_max[r] * (3.0f - sv) * 0.5f;
                if (l > 0.f) { hp_s += l; hp_c += 1.f; }
            }
            if ((mb & 8) && inc_hn[i0 + r]) {
                float l = s_min[r] * (1.0f - sv) * 0.5f - dv;
                if (l > 0.f) { hn_s += l; hn_c += 1.f; }
            }
        }
    }
    __syncthreads();

    // block-reduce the 4 partials, write per-(n, tile) slot (no atomics)
    float vals[4] = {hp_s, hp_c, hn_s, hn_c};
    for (int q = 0; q < 4; ++q) {
        sred[tid] = vals[q];
        __syncthreads();
        for (int s = 128; s > 0; s >>= 1) {
            if (tid < s) sred[tid] += sred[tid + s];
            __syncthreads();
        }
        if (tid == 0) part[((size_t)q * NB + n) * 32 + it] = sred[0];
        __syncthreads();
    }
}

// ---------------------------------------------------------------------------
// Kernel D: reduce tile partials, guarded means, final 4 outputs
// ---------------------------------------------------------------------------
__global__ __launch_bounds__(64)
void finalize_kernel(const float* __restrict__ part, float* __restrict__ out)
{
    int tid = threadIdx.x;   // 64 threads == NB
    float hp_s = 0.f, hp_c = 0.f, hn_s = 0.f, hn_c = 0.f;
    for (int it2 = 0; it2 < 32; ++it2) {
        hp_s += part[((size_t)0 * NB + tid) * 32 + it2];
        hp_c += part[((size_t)1 * NB + tid) * 32 + it2];
        hn_s += part[((size_t)2 * NB + tid) * 32 + it2];
        hn_c += part[((size_t)3 * NB + tid) * 32 + it2];
    }
    float hp_mean = (hp_c > 0.f) ? hp_s / hp_c : 0.f;
    float hn_mean = (hn_c > 0.f) ? hn_s / hn_c : 0.f;

    __shared__ float r0[64], r1[64], r2[64], r3[64];
    r0[tid] = hp_mean; r1[tid] = hn_mean; r2[tid] = hp_c; r3[tid] = hn_c;
    __syncthreads();
    for (int s = 32; s > 0; s >>= 1) {
        if (tid < s) {
            r0[tid] += r0[tid + s];
            r1[tid] += r1[tid + s];
            r2[tid] += r2[tid + s];
            r3[tid] += r3[tid + s];
        }
        __syncthreads();
    }
    if (tid == 0) {
        out[0] = r0[0] / (float)NB;
        out[1] = r1[0] / (float)NB;
        out[2] = r2[0] / (float)NB;
        out[3] = r3[0] / (float)NB;
    }
}

// ---------------------------------------------------------------------------
extern "C" void kernel_launch(void* const* d_in, const int* in_sizes, int n_in,
                              void* d_out, int out_size, void* d_ws, size_t ws_size,
                              hipStream_t stream)
{
    (void)in_sizes; (void)n_in; (void)out_size; (void)ws_size;
    const float* feat  = (const float*)d_in[0];
    const int*   label = (const int*)d_in[1];
    const int*   ltype = (const int*)d_in[2];
    const float* sim   = (const float*)d_in[3];
    float* out = (float*)d_out;

    char* ws = (char*)d_ws;
    unsigned char* mask    = (unsigned char*)(ws + OFF_MASK);
    int*           cnt     = (int*)(ws + OFF_CNT);
    unsigned char* ihp     = (unsigned char*)(ws + OFF_IHP);
    unsigned char* ihn     = (unsigned char*)(ws + OFF_IHN);
    float*         x2      = (float*)(ws + OFF_X2);
    float*         partial = (float*)(ws + OFF_PART);

    mask_kernel<<<M, 256, 0, stream>>>(label, ltype, sim, mask, cnt, ihp, ihn);
    x2_kernel<<<NB * M, 256, 0, stream>>>(feat, x2);
    dist_loss_kernel<<<dim3(32, NB), 256, 0, stream>>>(feat, sim, mask, cnt, ihp, ihn,
                                                       x2, partial);
    finalize_kernel<<<1, 64, 0, stream>>>(partial, out);
}